// GCN_2_13606456394529
// MI455X (gfx1250) — compile-verified
//
#include <hip/hip_runtime.h>
#include <hip/hip_bf16.h>

typedef __attribute__((ext_vector_type(2))) float v2f;
typedef __attribute__((ext_vector_type(8))) float v8f;

#define KDIM 128  // both GCN layers have K = 128

// ---------------------------------------------------------------------------
// Degree / normalization
// ---------------------------------------------------------------------------
__global__ void gcn_init_deg(float* __restrict__ deg, int n) {
    int i = blockIdx.x * blockDim.x + threadIdx.x;
    if (i < n) deg[i] = 1.0f;  // self-loop
}

__global__ void gcn_edge_deg(const long long* __restrict__ dst,
                             float* __restrict__ deg, int E) {
    int stride = gridDim.x * blockDim.x;
    for (int e = blockIdx.x * blockDim.x + threadIdx.x; e < E; e += stride) {
        int d = (int)dst[e];
        unsafeAtomicAdd(&deg[d], 1.0f);
    }
}

__global__ void gcn_dinv(const float* __restrict__ deg,
                         float* __restrict__ dinv, int n) {
    int i = blockIdx.x * blockDim.x + threadIdx.x;
    if (i < n) {
        float dg = deg[i];
        dinv[i] = (dg > 0.0f) ? __frsqrt_rn(dg) : 0.0f;
    }
}

// ---------------------------------------------------------------------------
// FP32 WMMA GEMM:  H[M x NOUT] = X[M x 128] * W[128 x NOUT]
// 256 threads = 8 waves; each wave owns one 16-row M tile.
// A frag (16x4 f32): lane L -> row = L&15; vgpr0 holds K = (L<16?0:2),
// vgpr1 holds K = (L<16?1:3).  B frag (4x16): same K split, col = L&15.
// C/D (16x16 f32): vgpr v -> row v (+8 for lanes 16-31), col = L&15.
// ---------------------------------------------------------------------------
template <int NOUT>
__global__ __launch_bounds__(256) void gcn_gemm_wmma(
    const float* __restrict__ X, const float* __restrict__ W,
    float* __restrict__ H, int M) {
    __shared__ float Wl[KDIM * NOUT];  // W staged row-major [k][n]
    for (int idx = threadIdx.x; idx < KDIM * NOUT; idx += 256)
        Wl[idx] = W[idx];
    __syncthreads();

    const int wave = threadIdx.x >> 5;
    const int lane = threadIdx.x & 31;
    const int mtiles = (M + 15) >> 4;
    const int mtile = blockIdx.x * 8 + wave;
    if (mtile >= mtiles) return;

    const int hi   = lane >> 4;       // 0: K pair {0,1}, 1: K pair {2,3}
    const int mrow = lane & 15;
    const int ncol = lane & 15;

    int row = mtile * 16 + mrow;
    int arow = row < M ? row : (M - 1);  // clamp A loads for ragged tail

    // Preload all 32 K-step A fragments (64 VGPRs).
    v2f a[KDIM / 4];
    const float* xr = X + (size_t)arow * KDIM + 2 * hi;
#pragma unroll
    for (int kk = 0; kk < KDIM / 4; ++kk) {
        a[kk].x = xr[4 * kk];
        a[kk].y = xr[4 * kk + 1];
    }

#pragma unroll
    for (int nt = 0; nt < NOUT / 16; ++nt) {
        v8f c = {};
        const int nbase = nt * 16 + ncol;
#pragma unroll
        for (int kk = 0; kk < KDIM / 4; ++kk) {
            v2f b;
            b.x = Wl[(4 * kk + 2 * hi) * NOUT + nbase];
            b.y = Wl[(4 * kk + 2 * hi + 1) * NOUT + nbase];
            c = __builtin_amdgcn_wmma_f32_16x16x4_f32(
                false, a[kk], false, b, (short)0, c, false, false);
        }
#pragma unroll
        for (int v = 0; v < 8; ++v) {
            int orow = mtile * 16 + v + 8 * hi;
            if (orow < M)
                H[(size_t)orow * NOUT + nbase] = c[v];
        }
    }
}

// ---------------------------------------------------------------------------
// agg[i] = h[i] * dinv[row]^2   (self-loop contribution, also inits buffer)
// ---------------------------------------------------------------------------
template <int F>
__global__ void gcn_init_agg(const float* __restrict__ h,
                             const float* __restrict__ dinv,
                             float* __restrict__ agg, int n) {
    int i = blockIdx.x * blockDim.x + threadIdx.x;
    int total = n * F;
    if (i >= total) return;
    float di = dinv[i / F];
    agg[i] = h[i] * di * di;
}

// ---------------------------------------------------------------------------
// Wave-per-edge scatter-add: agg[dst] += h[src] * dinv[src]*dinv[dst]
// F=128 -> float4/lane, F=64 -> float2/lane. Atomics land in L2 (51MB fits).
// ---------------------------------------------------------------------------
template <int F>
__global__ void gcn_edge_agg(const long long* __restrict__ src,
                             const long long* __restrict__ dst,
                             const float* __restrict__ dinv,
                             const float* __restrict__ h,
                             float* __restrict__ agg, int E) {
    const int V = F / 32;
    const int lane = threadIdx.x & 31;
    int wid = (blockIdx.x * blockDim.x + threadIdx.x) >> 5;
    int nw  = (gridDim.x * blockDim.x) >> 5;
    for (int e = wid; e < E; e += nw) {
        int s = (int)src[e];
        int d = (int)dst[e];
        float norm = dinv[s] * dinv[d];
        const float* hs = h + (size_t)s * F + lane * V;
        float* ad = agg + (size_t)d * F + lane * V;
        if (V == 4) {
            float4 hv = *(const float4*)hs;
            unsafeAtomicAdd(&ad[0], hv.x * norm);
            unsafeAtomicAdd(&ad[1], hv.y * norm);
            unsafeAtomicAdd(&ad[2], hv.z * norm);
            unsafeAtomicAdd(&ad[3], hv.w * norm);
        } else {
            float2 hv = *(const float2*)hs;
            unsafeAtomicAdd(&ad[0], hv.x * norm);
            unsafeAtomicAdd(&ad[1], hv.y * norm);
        }
    }
}

// ---------------------------------------------------------------------------
// In-place bias + ReLU over [n x 128]
// ---------------------------------------------------------------------------
__global__ void gcn_bias_relu(float* __restrict__ a,
                              const float* __restrict__ b, int total) {
    int i = blockIdx.x * blockDim.x + threadIdx.x;
    if (i >= total) return;
    float v = a[i] + b[i & 127];
    a[i] = v > 0.0f ? v : 0.0f;
}

// ---------------------------------------------------------------------------
// Wave-per-row bias + log_softmax over 64 features (wave32 shuffles)
// ---------------------------------------------------------------------------
__global__ void gcn_bias_logsoftmax(const float* __restrict__ agg,
                                    const float* __restrict__ b,
                                    float* __restrict__ out, int n) {
    const int lane = threadIdx.x & 31;
    int wid = (blockIdx.x * blockDim.x + threadIdx.x) >> 5;
    int nw  = (gridDim.x * blockDim.x) >> 5;
    for (int r = wid; r < n; r += nw) {
        float v0 = agg[(size_t)r * 64 + lane] + b[lane];
        float v1 = agg[(size_t)r * 64 + lane + 32] + b[lane + 32];
        float m = fmaxf(v0, v1);
#pragma unroll
        for (int off = 16; off > 0; off >>= 1)
            m = fmaxf(m, __shfl_xor(m, off, 32));
        float s = expf(v0 - m) + expf(v1 - m);
#pragma unroll
        for (int off = 16; off > 0; off >>= 1)
            s += __shfl_xor(s, off, 32);
        float lse = m + logf(s);
        out[(size_t)r * 64 + lane]      = v0 - lse;
        out[(size_t)r * 64 + lane + 32] = v1 - lse;
    }
}

// ---------------------------------------------------------------------------
// Launch: inputs = {x[N*128], edge_index[2*E] (int64), W1[128*128], b1[128],
//                   W2[128*64], b2[64]};  out = [N*64] f32
// ---------------------------------------------------------------------------
extern "C" void kernel_launch(void* const* d_in, const int* in_sizes, int n_in,
                              void* d_out, int out_size, void* d_ws, size_t ws_size,
                              hipStream_t stream) {
    const float* x  = (const float*)d_in[0];
    const long long* ei = (const long long*)d_in[1];
    const float* W1 = (const float*)d_in[2];
    const float* b1 = (const float*)d_in[3];
    const float* W2 = (const float*)d_in[4];
    const float* b2 = (const float*)d_in[5];
    float* out = (float*)d_out;

    const int N = in_sizes[0] / 128;
    const int E = in_sizes[1] / 2;
    const long long* src = ei;
    const long long* dst = ei + E;

    // Workspace layout (floats): deg | dinv | h1(N*128) | agg1(N*128) | agg2(N*64)
    float* deg  = (float*)d_ws;
    float* dinv = deg + N;
    float* h1   = dinv + N;                 // also reused as h2 (N*64 <= N*128)
    float* agg1 = h1 + (size_t)N * 128;
    float* agg2 = agg1 + (size_t)N * 128;

    const int T = 256;
    const int mtiles = (N + 15) / 16;
    const int gemmBlocks = (mtiles + 7) / 8;
    const int edgeBlocks = 16384;

    // Normalization
    gcn_init_deg<<<(N + T - 1) / T, T, 0, stream>>>(deg, N);
    gcn_edge_deg<<<edgeBlocks, T, 0, stream>>>(dst, deg, E);
    gcn_dinv<<<(N + T - 1) / T, T, 0, stream>>>(deg, dinv, N);

    // Layer 1: h1 = x @ W1; agg1 = scatter(norm * h1[src]) + selfloop; relu(+b1)
    gcn_gemm_wmma<128><<<gemmBlocks, T, 0, stream>>>(x, W1, h1, N);
    gcn_init_agg<128><<<((size_t)N * 128 + T - 1) / T, T, 0, stream>>>(h1, dinv, agg1, N);
    gcn_edge_agg<128><<<edgeBlocks, T, 0, stream>>>(src, dst, dinv, h1, agg1, E);
    gcn_bias_relu<<<((size_t)N * 128 + T - 1) / T, T, 0, stream>>>(agg1, b1, N * 128);

    // Layer 2: h2 = relu1 @ W2 (into h1 buffer); agg2; bias + log_softmax
    gcn_gemm_wmma<64><<<gemmBlocks, T, 0, stream>>>(agg1, W2, h1, N);
    gcn_init_agg<64><<<((size_t)N * 64 + T - 1) / T, T, 0, stream>>>(h1, dinv, agg2, N);
    gcn_edge_agg<64><<<edgeBlocks, T, 0, stream>>>(src, dst, dinv, h1, agg2, E);
    gcn_bias_logsoftmax<<<(N / 8) + 1, T, 0, stream>>>(agg2, b2, out, N);
}